// fitparams_81097572483405
// MI455X (gfx1250) — compile-verified
//
#include <hip/hip_runtime.h>

// PROSPECT-D on MI455X (gfx1250).
//
// Structure:
//   K1: per-wavelength _getav constants (talf, t12, t21) -> d_ws   [2101 threads]
//   K2: one wave32 per 16x16 (sample x wavelength) tile.
//       kall tile = 2x V_WMMA_F32_16X16X4_F32 (traits/N x spectral coefs),
//       then the transcendental chain runs in-register on the 8 f32
//       accumulators using the documented C/D layout:
//         VGPR i, lane L  ->  M = i + 8*(L>=16), Ncol = L%16.
//       Fast HW transcendentals (v_exp/v_log/v_rcp/v_sqrt_f32) keep the
//       ~50-op/element chain near the ~6us HBM store floor (138 MB @ 23.3TB/s).

#define NLAM 2101
#define BSAMP 8192

typedef __attribute__((ext_vector_type(2))) float v2f;
typedef __attribute__((ext_vector_type(8))) float v8f;

__device__ __forceinline__ float frcp(float x)  { return __builtin_amdgcn_rcpf(x); }
__device__ __forceinline__ float fsqrt(float x) { return __builtin_amdgcn_sqrtf(x); }
__device__ __forceinline__ float fexp(float x)  { return __builtin_amdgcn_exp2f(x * 1.4426950408889634f); }
__device__ __forceinline__ float flog(float x)  { return __builtin_amdgcn_logf(x) * 0.6931471805599453f; }
__device__ __forceinline__ float fpow(float b, float e) {
    return __builtin_amdgcn_exp2f(e * __builtin_amdgcn_logf(b));
}

// Stern transmissivity (PROSPECT 'tav'); only depends on nr -> precomputed per l.
__device__ float getav(float sa2, bool deg90, float nr) {
    float n2  = nr * nr;
    float npx = n2 + 1.0f;
    float nm2 = (n2 - 1.0f) * (n2 - 1.0f);
    float a   = (nr + 1.0f) * (nr + 1.0f) * 0.5f;
    float k   = -nm2 * 0.25f;
    float b2  = sa2 - npx * 0.5f;
    float b1  = deg90 ? 0.0f : sqrtf(b2 * b2 + k);
    float b   = b1 - b2;
    float k2  = k * k;
    float ts  = (k2 / (6.0f * b * b * b) + k / b - b * 0.5f)
              - (k2 / (6.0f * a * a * a) + k / a - a * 0.5f);
    float tp1 = -2.0f * n2 * (b - a) / (npx * npx);
    float tp2 = -2.0f * n2 * npx * logf(b / a) / nm2;
    float tp3 = n2 * (1.0f / b - 1.0f / a) * 0.5f;
    float n22 = n2 * n2;
    float npx3 = npx * npx * npx;
    float npax_a = 2.0f * npx * a - nm2;
    float npx_b  = 2.0f * npx * b - nm2;
    float tp4 = 16.0f * n22 * (n22 + 1.0f) * logf(npx_b / npax_a) / (npx3 * nm2);
    float tp5 = 16.0f * (n2 * n2 * n2) * (1.0f / npx_b - 1.0f / npax_a) / npx3;
    return (ts + tp1 + tp2 + tp3 + tp4 + tp5) / (2.0f * sa2);
}

__global__ void prospect_spectral(const float* __restrict__ nr, float* __restrict__ spec) {
    int l = blockIdx.x * blockDim.x + threadIdx.x;
    if (l >= NLAM) return;
    float n = nr[l];
    float s40 = sinf(0.6981317007977318f);           // 40 degrees
    float talf = getav(s40 * s40, false, n);
    float t12  = getav(1.0f, true, n);               // sin(90)^2 == 1, b1 == 0
    spec[l]             = talf;
    spec[NLAM + l]      = t12;
    spec[2 * NLAM + l]  = t12 / (n * n);             // t21
}

__global__ void __launch_bounds__(256) prospect_main(
    const float* __restrict__ Np,   const float* __restrict__ cab,
    const float* __restrict__ car,  const float* __restrict__ water,
    const float* __restrict__ lma,  const float* __restrict__ cant,
    const float* __restrict__ kab,  const float* __restrict__ kcar,
    const float* __restrict__ kant, const float* __restrict__ kw,
    const float* __restrict__ km,   const float* __restrict__ spec,
    float* __restrict__ out)
{
    const int NT = (NLAM + 15) >> 4;                        // 132 column tiles
    int wave   = blockIdx.x * (blockDim.x >> 5) + (threadIdx.x >> 5);
    int tile_n = wave % NT;
    int tile_m = wave / NT;
    if (tile_m >= (BSAMP >> 4)) return;                     // wave-uniform guard

    int lane = threadIdx.x & 31;
    int l16  = lane & 15;
    int half = lane >> 4;                                   // 0: lanes 0-15, 1: lanes 16-31

    int m   = tile_m * 16 + l16;                            // A row owned by this lane
    int col = tile_n * 16 + l16;                            // B column owned by this lane
    int lc  = col < NLAM ? col : NLAM - 1;                  // clamp ragged tail for loads
    bool colok = col < NLAM;

    // ---- A matrix: 16x4 f32, lane holds M=lane%16; V0 = K0|K2, V1 = K1|K3 per half.
    //      K rows (wmma #1): cab, car, cant, water; (wmma #2): lma, 0, 0, 0.
    //      1/N folded into A so the accumulator is kall directly.
    float invN = frcp(Np[m]);
    v2f aLo, aHi;
    aLo.x = (half ? cant[m]  : cab[m]) * invN;              // K=0 | K=2
    aLo.y = (half ? water[m] : car[m]) * invN;              // K=1 | K=3
    aHi.x = half ? 0.0f : lma[m] * invN;                    // K=4 | K=6(=0)
    aHi.y = 0.0f;                                           // K=5 | K=7(=0)

    // ---- B matrix: 4x16 f32, lane holds N=lane%16; V0 = K0|K2, V1 = K1|K3 per half.
    v2f bLo, bHi;
    bLo.x = half ? kant[lc] : kab[lc];
    bLo.y = half ? kw[lc]   : kcar[lc];
    bHi.x = half ? 0.0f     : km[lc];
    bHi.y = 0.0f;

    v8f acc = {0.f, 0.f, 0.f, 0.f, 0.f, 0.f, 0.f, 0.f};
    acc = __builtin_amdgcn_wmma_f32_16x16x4_f32(false, aLo, false, bLo, (short)0, acc, false, false);
    acc = __builtin_amdgcn_wmma_f32_16x16x4_f32(false, aHi, false, bHi, (short)0, acc, false, false);

    // Per-column spectral constants (fixed per lane).
    float talf = spec[lc];
    float t12v = spec[NLAM + lc];
    float t21v = spec[2 * NLAM + lc];
    float ralf = 1.0f - talf;
    float r12v = 1.0f - t12v;
    float r21v = 1.0f - t21v;

    // Per-row (N-1): rows of this lane's accumulators are mbase..mbase+7 (32B aligned).
    int mbase = tile_m * 16 + half * 8;
    const float4* np4 = reinterpret_cast<const float4*>(Np + mbase);
    float4 nA = np4[0];
    float4 nB = np4[1];
    float nm1[8] = { nA.x - 1.f, nA.y - 1.f, nA.z - 1.f, nA.w - 1.f,
                     nB.x - 1.f, nB.y - 1.f, nB.z - 1.f, nB.w - 1.f };

    float* reflp = out;
    float* tranp = out + (size_t)BSAMP * NLAM;

#pragma unroll
    for (int i = 0; i < 8; ++i) {
        float kall = acc[i] > 1e-4f ? acc[i] : 1e-4f;
        float em = fexp(-kall);

        // E1(kall) = -expi(-kall): A&S 5.1.53 (x<=1), 5.1.56 (x>1).
        float x  = kall;
        float e1s = ((((0.00107857f * x - 0.00976004f) * x + 0.05519968f) * x
                      - 0.24991055f) * x + 0.99999193f) * x - 0.57721566f - flog(x);
        float pn = (((x + 8.5733287401f) * x + 18.059016973f) * x + 8.6347608925f) * x + 0.2677737343f;
        float pd = (((x + 9.5733223454f) * x + 25.6329561486f) * x + 21.0996530827f) * x + 3.9584969228f;
        float e1l = em * pn * frcp(pd * x);
        float e1  = x > 1.0f ? e1l : e1s;
        float tau = (1.0f - kall) * em + kall * kall * e1;

        // one-layer refl/trans
        float tt21 = tau * t21v;
        float rd0  = frcp(1.0f - r21v * r21v * tau * tau);
        float Ta = talf * tt21 * rd0;
        float Ra = ralf + r21v * tau * Ta;
        float t  = t12v * tt21 * rd0;
        float r  = r12v + r21v * tau * t;

        // N-1 interior layers (Stokes)
        float rpt = r + t, rmt = r - t;
        float D  = fsqrt((1.0f + rpt) * (1.0f + rmt) * (1.0f - rmt) * (1.0f - rpt));
        float rq = r * r - t * t;
        float aa = (1.0f + rq + D) * frcp(2.0f * r);
        float bb = (1.0f - rq + D) * frcp(2.0f * t);
        float e  = nm1[i];
        float bnm1 = fpow(bb, e);
        float bn2  = bnm1 * bnm1;
        float a2   = aa * aa;
        float dn   = frcp(a2 * bn2 - 1.0f);
        float Rsub = aa * (bn2 - 1.0f) * dn;
        float Tsub = bnm1 * (a2 - 1.0f) * dn;

        // conservative-scattering branch
        float Tsat = t * frcp(t + (1.0f - t) * e);
        bool  sat  = rpt >= 1.0f;
        Tsub = sat ? Tsat : Tsub;
        Rsub = sat ? 1.0f - Tsat : Rsub;

        float rd2  = frcp(1.0f - Rsub * r);
        float tran = Ta * Tsub * rd2;
        float refl = Ra + Ta * Rsub * t * rd2;

        if (colok) {
            size_t row = (size_t)(mbase + i);
            reflp[row * NLAM + col] = refl;
            tranp[row * NLAM + col] = tran;
        }
    }
}

extern "C" void kernel_launch(void* const* d_in, const int* in_sizes, int n_in,
                              void* d_out, int out_size, void* d_ws, size_t ws_size,
                              hipStream_t stream) {
    // setup_inputs order: N, cab, car, water, lma, cant, nr, kab, kcar, kant, kw, km
    const float* Np    = (const float*)d_in[0];
    const float* cab   = (const float*)d_in[1];
    const float* car   = (const float*)d_in[2];
    const float* water = (const float*)d_in[3];
    const float* lma   = (const float*)d_in[4];
    const float* cant  = (const float*)d_in[5];
    const float* nr    = (const float*)d_in[6];
    const float* kab   = (const float*)d_in[7];
    const float* kcar  = (const float*)d_in[8];
    const float* kant  = (const float*)d_in[9];
    const float* kw    = (const float*)d_in[10];
    const float* km    = (const float*)d_in[11];
    float* out  = (float*)d_out;
    float* spec = (float*)d_ws;                   // 3*NLAM floats

    prospect_spectral<<<(NLAM + 255) / 256, 256, 0, stream>>>(nr, spec);

    int tilesM = BSAMP >> 4;                      // 512
    int tilesN = (NLAM + 15) >> 4;                // 132
    int waves  = tilesM * tilesN;                 // 67584
    int blocks = (waves + 7) / 8;                 // 8 waves (256 thr) per block
    prospect_main<<<blocks, 256, 0, stream>>>(Np, cab, car, water, lma, cant,
                                              kab, kcar, kant, kw, km, spec, out);
}